// BoxLoss_54382875902460
// MI455X (gfx1250) — compile-verified
//
#include <hip/hip_runtime.h>
#include <hip/hip_bf16.h>
#include <math.h>

// CDNA5 wave32 vector types for WMMA f32 16x16x4:
// A (16x4 f32) = 64 elems / 32 lanes = 2 VGPRs -> v2f
// B (4x16 f32) = 64 elems / 32 lanes = 2 VGPRs -> v2f
// C/D (16x16 f32) = 256 elems / 32 lanes = 8 VGPRs -> v8f
typedef __attribute__((ext_vector_type(2))) float v2f;
typedef __attribute__((ext_vector_type(8))) float v8f;

#define NTGT 50
#define NIMG 32
#define NA   3

// One block per (scale, image) unit: 3*32 = 96 blocks, 64 threads each.
// Sparse path: never touches the dense (A,H,W,4) gt tensor.
__global__ void box_loss_partial(const float* __restrict__ out0,
                                 const float* __restrict__ anc0,
                                 const float* __restrict__ out1,
                                 const float* __restrict__ anc1,
                                 const float* __restrict__ out2,
                                 const float* __restrict__ anc2,
                                 const float* __restrict__ tgt,
                                 float* __restrict__ partial)
{
    const int unit = blockIdx.x;          // 0..95
    const int s    = unit / NIMG;         // scale 0..2
    const int b    = unit % NIMG;         // image 0..31
    const int tid  = threadIdx.x;         // 0..63

    const float* outp = (s == 0) ? out0 : (s == 1) ? out1 : out2;
    const float* ancp = (s == 0) ? anc0 : (s == 1) ? anc1 : anc2;
    const int G = (s == 0) ? 52 : (s == 1) ? 26 : 13;
    const int H = G, W = G;

    __shared__ float s_tx[NTGT], s_ty[NTGT], s_tw[NTGT], s_th[NTGT];
    __shared__ int   s_cell[NTGT];
    __shared__ float s_sum[64];
    __shared__ int   s_cnt[64];

    if (tid < NTGT) {
        const float* r = tgt + ((size_t)b * NTGT + tid) * 5;
        float rx = r[1], ry = r[2], rw = r[3], rh = r[4];
        bool valid = !(rx == 0.f && ry == 0.f && rw == 0.f && rh == 0.f);

        float tx = rx * (float)W, ty = ry * (float)H;
        float tw = rw * (float)W, th = rh * (float)H;
        float cx = floorf(tx), cy = floorf(ty);
        float zx = tx - cx - 0.5f, zy = ty - cy - 0.5f;

        // target rect (centered at (zx,zy))
        float tx0 = zx - tw * 0.5f, ty0 = zy - th * 0.5f;
        float tx1 = zx + tw * 0.5f, ty1 = zy + th * 0.5f;
        float area_t = (tx1 - tx0) * (ty1 - ty0);

        float best = -1.0f; int banc = 0;
#pragma unroll
        for (int a = 0; a < NA; ++a) {
            float aw = ancp[a * 2 + 0], ah = ancp[a * 2 + 1];
            float ax0 = -aw * 0.5f, ay0 = -ah * 0.5f;
            float ax1 =  aw * 0.5f, ay1 =  ah * 0.5f;
            float x0 = fmaxf(tx0, ax0), y0 = fmaxf(ty0, ay0);
            float x1 = fminf(tx1, ax1), y1 = fminf(ty1, ay1);
            float flag = (x0 < x1 && y0 < y1) ? 1.0f : 0.0f;
            float inter = (x1 - x0) * (y1 - y0) * flag;
            float area_a = (ax1 - ax0) * (ay1 - ay0);
            float iou = inter / (area_t + area_a - inter);
            if (iou > best) { best = iou; banc = a; }   // first-max wins (argmax)
        }

        int icx = (int)cx, icy = (int)cy;
        bool kept = valid && (best > 0.5f) &&
                    icx >= 0 && icx < W && icy >= 0 && icy < H;   // scatter mode='drop'
        s_cell[tid] = kept ? ((banc * H + icy) * W + icx) : -1;
        s_tx[tid] = tx; s_ty[tid] = ty; s_tw[tid] = tw; s_th[tid] = th;
    }
    __syncthreads();

    float acc = 0.f; int cnt = 0;
    if (tid < NTGT) {
        int cell = s_cell[tid];
        bool survive = (cell >= 0);
        if (survive) {
            // scatter-set collision: later target overwrites earlier (XLA order)
            for (int j = tid + 1; j < NTGT; ++j)
                if (s_cell[j] == cell) { survive = false; break; }
        }
        if (survive) {
            size_t base = ((size_t)b * (NA * H * W) + (size_t)cell) * 85;
            float px = outp[base + 0], py = outp[base + 1];
            float pw = outp[base + 2], ph = outp[base + 3];
            float gx = s_tx[tid], gy = s_ty[tid];
            float gw = s_tw[tid], gh = s_th[tid];
            float dx = px - gx, dy = py - gy;
            float dw = 1.0f / sqrtf(pw) - 1.0f / sqrtf(gw);
            float dh = 1.0f / sqrtf(ph) - 1.0f / sqrtf(gh);
            acc = dx * dx + dy * dy + dw * dw + dh * dh;
            cnt = 1;
        }
    }
    s_sum[tid] = acc; s_cnt[tid] = cnt;
    __syncthreads();

    // fixed-order tree reduction over 64 threads (deterministic fp)
    for (int off = 32; off > 0; off >>= 1) {
        if (tid < off) {
            s_sum[tid] += s_sum[tid + off];
            s_cnt[tid] += s_cnt[tid + off];
        }
        __syncthreads();
    }
    if (tid == 0) {
        int n = s_cnt[0];
        partial[unit] = (n > 0) ? s_sum[0] / (2.0f * (float)n) : 0.0f;
    }
}

// Final 96 -> 1 reduction: 32 lanes each fold 3 partials, then one
// V_WMMA_F32_16X16X4_F32 against a ones B-matrix produces row sums
// D[m][n] = v[m] + v[m+16] (fp32-exact path: only 2 nonzero products/row).
// Column 0 lives in lane n (M=0..7) and lane n+16 (M=8..15), so an
// in-lane 8-add + shfl_xor(16) completes the 32-lane sum.
__global__ void box_loss_reduce(const float* __restrict__ partial,
                                float* __restrict__ out)
{
    const int lane = threadIdx.x;   // exactly 32 threads, EXEC all-1s for WMMA
    float v = partial[lane] + partial[lane + 32] + partial[lane + 64];

    v2f A;  A.x = v;    A.y = 0.0f;   // A[m][0]=v(lane m), A[m][2]=v(lane m+16)
    v2f Bm; Bm.x = 1.0f; Bm.y = 1.0f; // all-ones B regardless of lane layout
    v8f C = {};
    v8f D = __builtin_amdgcn_wmma_f32_16x16x4_f32(
        /*neg_a=*/false, A, /*neg_b=*/false, Bm,
        /*c_mod=*/(short)0, C, /*reuse_a=*/false, /*reuse_b=*/false);

    float ssum = D[0] + D[1] + D[2] + D[3] + D[4] + D[5] + D[6] + D[7];
    float tot  = ssum + __shfl_xor(ssum, 16, 32);
    if (lane == 0) out[0] = tot * (1.0f / 32.0f);   // divide by B
}

extern "C" void kernel_launch(void* const* d_in, const int* in_sizes, int n_in,
                              void* d_out, int out_size, void* d_ws, size_t ws_size,
                              hipStream_t stream) {
    const float* out0 = (const float*)d_in[0];
    const float* anc0 = (const float*)d_in[1];
    const float* out1 = (const float*)d_in[2];
    const float* anc1 = (const float*)d_in[3];
    const float* out2 = (const float*)d_in[4];
    const float* anc2 = (const float*)d_in[5];
    const float* tgt  = (const float*)d_in[6];

    float* partial = (float*)d_ws;   // 96 floats of scratch

    box_loss_partial<<<96, 64, 0, stream>>>(out0, anc0, out1, anc1,
                                            out2, anc2, tgt, partial);
    box_loss_reduce<<<1, 32, 0, stream>>>(partial, (float*)d_out);
}